// ProposalLayer_23519240913470
// MI455X (gfx1250) — compile-verified
//
#include <hip/hip_runtime.h>
#include <hip/hip_bf16.h>
#include <cstdint>

// ---------------- problem constants (match reference) ----------------
constexpr int B        = 4;
constexpr int A        = 12;
constexpr int FH       = 50;
constexpr int FW       = 84;
constexpr int NANCH    = A * FH * FW;     // 50400 anchors per batch
constexpr int SORT_N   = 65536;           // next pow2 >= NANCH
constexpr int PRE_NMS  = 6000;
constexpr int POST_NMS = 300;
constexpr int MASK_WORDS = (PRE_NMS + 31) / 32;   // 188
constexpr float NMS_THRESH = 0.7f;
constexpr float MIN_SIZE   = 16.0f;
constexpr float NEG_INF    = -3.0e38f;

// ---------------- CDNA5 async global->LDS gather helpers ----------------
__device__ __forceinline__ void async_gather_b128_to_lds(uint32_t lds_off,
                                                         const void* gptr) {
  // gfx1250: copies 16 bytes from global memory to LDS, per-lane addresses,
  // tracked with ASYNCcnt.
  asm volatile("global_load_async_to_lds_b128 %0, %1, off"
               :
               : "v"(lds_off), "v"((unsigned long long)(uintptr_t)gptr)
               : "memory");
}
__device__ __forceinline__ void wait_asynccnt0() {
  asm volatile("s_wait_asynccnt 0" ::: "memory");
}

// ---------------- 1) decode: softmax + bbox transform + clip ----------------
__global__ void decode_kernel(const float* __restrict__ scores,
                              const float* __restrict__ deltas,
                              const float* __restrict__ anchors,
                              const float* __restrict__ im_info,
                              float* __restrict__ keys,
                              int*   __restrict__ idxs,
                              float* __restrict__ boxes) {
  int t = blockIdx.x * blockDim.x + threadIdx.x;
  if (t >= B * SORT_N) return;
  int b = t / SORT_N;
  int e = t - b * SORT_N;
  if (e >= NANCH) {            // pad for pow2 bitonic sort
    keys[t] = NEG_INF;
    idxs[t] = -1;
    return;
  }
  int loc = e / A, a = e - loc * A;
  int y = loc / FW, x = loc - y * FW;
  const int HW = FH * FW;

  int sbase = (b * (2 * A)) * HW + y * FW + x;
  int dbase = (b * (4 * A)) * HW + y * FW + x;
  __builtin_prefetch(deltas + dbase + (4 * a) * HW, 0, 0);  // global_prefetch_b8

  float s0 = scores[sbase + (2 * a + 0) * HW];
  float s1 = scores[sbase + (2 * a + 1) * HW];
  float p  = 1.0f / (1.0f + __expf(s0 - s1));   // softmax fg prob

  float d0 = deltas[dbase + (4 * a + 0) * HW];
  float d1 = deltas[dbase + (4 * a + 1) * HW];
  float d2 = deltas[dbase + (4 * a + 2) * HW];
  float d3 = deltas[dbase + (4 * a + 3) * HW];

  const float* an = anchors + (size_t)e * 4;
  float ax1 = an[0], ay1 = an[1], ax2 = an[2], ay2 = an[3];
  float w  = ax2 - ax1 + 1.0f;
  float h  = ay2 - ay1 + 1.0f;
  float cx = ax1 + 0.5f * w;
  float cy = ay1 + 0.5f * h;
  float pcx = d0 * w + cx;
  float pcy = d1 * h + cy;
  float pw  = __expf(d2) * w;
  float ph  = __expf(d3) * h;

  float imh = im_info[b * 4 + 0];
  float imw = im_info[b * 4 + 1];
  float sc2 = im_info[b * 4 + 2];
  float sc3 = im_info[b * 4 + 3];

  float bx1 = fminf(fmaxf(pcx - 0.5f * pw, 0.0f), imw - 1.0f);
  float by1 = fminf(fmaxf(pcy - 0.5f * ph, 0.0f), imh - 1.0f);
  float bx2 = fminf(fmaxf(pcx + 0.5f * pw, 0.0f), imw - 1.0f);
  float by2 = fminf(fmaxf(pcy + 0.5f * ph, 0.0f), imh - 1.0f);

  float ws = bx2 - bx1 + 1.0f;
  float hs = by2 - by1 + 1.0f;
  bool valid = (ws >= MIN_SIZE * sc3) && (hs >= MIN_SIZE * sc2);

  float* bx = boxes + ((size_t)b * NANCH + e) * 4;
  bx[0] = bx1; bx[1] = by1; bx[2] = bx2; bx[3] = by2;
  keys[t] = valid ? p : -1.0f;
  idxs[t] = e;
}

// ---------------- 2) hybrid bitonic sort (descending), key+payload ----------
// Chunk = 1024 elements in LDS, 512 threads/block.
__device__ __forceinline__ void cmp_swap_lds(float* sk, int* si, int i, int p,
                                             bool desc) {
  float a = sk[i], bv = sk[p];
  bool sw = desc ? (a < bv) : (a > bv);
  if (sw) {
    sk[i] = bv; sk[p] = a;
    int ta = si[i]; si[i] = si[p]; si[p] = ta;
  }
}

__global__ void bitonic_local_sort(float* __restrict__ keys,
                                   int* __restrict__ idxs) {
  __shared__ float sk[1024];
  __shared__ int   si[1024];
  const int base = blockIdx.x * 1024;         // flat over B*SORT_N
  const int gb   = base & (SORT_N - 1);       // in-batch element base
  const int t = threadIdx.x;                  // 512 threads
  sk[t]       = keys[base + t];       si[t]       = idxs[base + t];
  sk[t + 512] = keys[base + t + 512]; si[t + 512] = idxs[base + t + 512];
  __syncthreads();
  for (int k = 2; k <= 1024; k <<= 1) {
    for (int j = k >> 1; j >= 1; j >>= 1) {
      int i = ((t & ~(j - 1)) << 1) | (t & (j - 1));
      int p = i | j;
      bool desc = (((gb + i) & k) == 0);
      cmp_swap_lds(sk, si, i, p, desc);
      __syncthreads();
    }
  }
  keys[base + t]       = sk[t];       idxs[base + t]       = si[t];
  keys[base + t + 512] = sk[t + 512]; idxs[base + t + 512] = si[t + 512];
}

__global__ void bitonic_global_step(float* __restrict__ keys,
                                    int* __restrict__ idxs, int k, int j) {
  int t = blockIdx.x * blockDim.x + threadIdx.x;     // B * SORT_N/2 threads
  if (t >= B * (SORT_N / 2)) return;
  int b = t / (SORT_N / 2);
  int e = t - b * (SORT_N / 2);
  int i = ((e & ~(j - 1)) << 1) | (e & (j - 1));
  int p = i | j;
  bool desc = ((i & k) == 0);
  int base = b * SORT_N;
  float a = keys[base + i], bv = keys[base + p];
  bool sw = desc ? (a < bv) : (a > bv);
  if (sw) {
    keys[base + i] = bv; keys[base + p] = a;
    int ta = idxs[base + i]; idxs[base + i] = idxs[base + p]; idxs[base + p] = ta;
  }
}

__global__ void bitonic_local_merge(float* __restrict__ keys,
                                    int* __restrict__ idxs, int k) {
  __shared__ float sk[1024];
  __shared__ int   si[1024];
  const int base = blockIdx.x * 1024;
  const int gb   = base & (SORT_N - 1);
  const int t = threadIdx.x;                  // 512 threads
  sk[t]       = keys[base + t];       si[t]       = idxs[base + t];
  sk[t + 512] = keys[base + t + 512]; si[t + 512] = idxs[base + t + 512];
  __syncthreads();
  for (int j = 512; j >= 1; j >>= 1) {
    int i = ((t & ~(j - 1)) << 1) | (t & (j - 1));
    int p = i | j;
    bool desc = (((gb + i) & k) == 0);
    cmp_swap_lds(sk, si, i, p, desc);
    __syncthreads();
  }
  keys[base + t]       = sk[t];       idxs[base + t]       = si[t];
  keys[base + t + 512] = sk[t + 512]; idxs[base + t + 512] = si[t + 512];
}

// ---------------- 3) NMS suppression bitmask (256 rows x 32 cols / block) ----
__global__ void nms_mask_kernel(const int* __restrict__ idxs,
                                const float* __restrict__ boxes,
                                unsigned* __restrict__ mask) {
  const int b = blockIdx.z;
  const int rowStart = blockIdx.x * blockDim.x;
  const int row = rowStart + threadIdx.x;
  const int cb = blockIdx.y;
  const int colBase = cb * 32;

  if (colBase + 31 <= rowStart) {            // strict upper triangle only
    if (row < PRE_NMS)
      mask[((size_t)b * PRE_NMS + row) * MASK_WORDS + cb] = 0u;
    return;
  }

  __shared__ float4 cbox[32];
  // Wave 0 gathers the 32 column boxes (indirect through sorted order) into
  // LDS via the CDNA5 async DMA path, waits on its own ASYNCcnt, then the
  // block barrier publishes the tile to the other waves.
  if (threadIdx.x < 32) {
    int l = threadIdx.x;
    int c = colBase + l;
    int ci = (c < PRE_NMS) ? idxs[b * SORT_N + c] : 0;
    if (ci < 0) ci = 0;
    const float* src = boxes + ((size_t)b * NANCH + ci) * 4;
    uint32_t lds_off = (uint32_t)(uintptr_t)(&cbox[l]);
    async_gather_b128_to_lds(lds_off, src);
    wait_asynccnt0();
  }
  __syncthreads();

  if (row >= PRE_NMS) return;
  int ri = idxs[b * SORT_N + row];
  if (ri < 0) ri = 0;
  const float4 rb = *(const float4*)(boxes + ((size_t)b * NANCH + ri) * 4);
  float rx1 = rb.x, ry1 = rb.y, rx2 = rb.z, ry2 = rb.w;
  float rarea = (rx2 - rx1 + 1.0f) * (ry2 - ry1 + 1.0f);

  unsigned bits = 0u;
#pragma unroll
  for (int l = 0; l < 32; ++l) {
    float4 c4 = cbox[l];                     // ds_load_b128
    float xx1 = fmaxf(rx1, c4.x), yy1 = fmaxf(ry1, c4.y);
    float xx2 = fminf(rx2, c4.z), yy2 = fminf(ry2, c4.w);
    float iw = fmaxf(xx2 - xx1 + 1.0f, 0.0f);
    float ih = fmaxf(yy2 - yy1 + 1.0f, 0.0f);
    float inter = iw * ih;
    float carea = (c4.z - c4.x + 1.0f) * (c4.w - c4.y + 1.0f);
    float iou = inter / (rarea + carea - inter);
    int c = colBase + l;
    if (c > row && c < PRE_NMS && iou > NMS_THRESH) bits |= (1u << l);
  }
  mask[((size_t)b * PRE_NMS + row) * MASK_WORDS + cb] = bits;
}

// ---------------- 4) greedy scan, one wave32 per batch -----------------------
// Keys are sorted descending, so valid entries (score in (0,1)) form a prefix;
// count it with ballots, then walk live words survivor-to-survivor with ffs.
// ~(#keeps + #words) iterations instead of PRE_NMS.
__global__ void nms_scan_kernel(const float* __restrict__ keys,
                                const unsigned* __restrict__ mask,
                                int* __restrict__ keepPos,
                                int* __restrict__ keepCnt) {
  const int b = blockIdx.x;
  const int lane = threadIdx.x;              // 32 threads = 1 wave
  __shared__ unsigned removed[MASK_WORDS];
  for (int w = lane; w < MASK_WORDS; w += 32) removed[w] = 0u;
  __syncthreads();

  // number of valid (non-suppressible-by-construction) sorted entries
  int nValid = 0;
  for (int basei = 0; basei < PRE_NMS; basei += 32) {
    int i = basei + lane;
    bool ok = (i < PRE_NMS) && (keys[b * SORT_N + i] > -0.5f);
    unsigned long long m = __ballot(ok);
    nValid += __popcll(m);
  }

  int cnt = 0;                               // identical across all lanes
  for (int w = 0; w < MASK_WORDS; ++w) {
    int basei = w * 32;
    if (basei >= nValid || cnt >= POST_NMS) break;
    unsigned validMask = 0xFFFFFFFFu;
    int rem = nValid - basei;
    if (rem < 32) validMask = (1u << rem) - 1u;
    unsigned live = ~removed[w] & validMask;
    while (live != 0u && cnt < POST_NMS) {
      int l = __ffs(live) - 1;               // lowest surviving rank in word
      int i = basei + l;
      if (lane == 0) keepPos[b * POST_NMS + cnt] = i;
      ++cnt;
      const unsigned* mrow = mask + ((size_t)b * PRE_NMS + i) * MASK_WORDS;
      for (int ww = lane; ww < MASK_WORDS; ww += 32) removed[ww] |= mrow[ww];
      __syncthreads();
      live &= ~(1u << l);
      live &= ~removed[w];                   // re-read updated word
    }
  }
  if (lane == 0) keepCnt[b] = cnt;
}

// ---------------- 5) emit (b, x1, y1, x2, y2), zero-filled tail -------------
__global__ void write_out_kernel(const int* __restrict__ idxs,
                                 const float* __restrict__ boxes,
                                 const int* __restrict__ keepPos,
                                 const int* __restrict__ keepCnt,
                                 float* __restrict__ out) {
  int t = blockIdx.x * blockDim.x + threadIdx.x;
  if (t >= B * POST_NMS) return;
  int b = t / POST_NMS;
  int s = t - b * POST_NMS;
  float x1 = 0.0f, y1 = 0.0f, x2 = 0.0f, y2 = 0.0f;
  if (s < keepCnt[b]) {
    int i  = keepPos[b * POST_NMS + s];
    int oi = idxs[b * SORT_N + i];
    if (oi >= 0) {
      const float* bx = boxes + ((size_t)b * NANCH + oi) * 4;
      x1 = bx[0]; y1 = bx[1]; x2 = bx[2]; y2 = bx[3];
    }
  }
  float* o = out + (size_t)t * 5;
  o[0] = (float)b; o[1] = x1; o[2] = y1; o[3] = x2; o[4] = y2;
}

// ---------------- launch ----------------------------------------------------
extern "C" void kernel_launch(void* const* d_in, const int* in_sizes, int n_in,
                              void* d_out, int out_size, void* d_ws,
                              size_t ws_size, hipStream_t stream) {
  const float* scores  = (const float*)d_in[0];
  const float* deltas  = (const float*)d_in[1];
  const float* anchors = (const float*)d_in[2];
  const float* im_info = (const float*)d_in[3];
  float* out = (float*)d_out;

  // workspace layout (all 16B aligned)
  char* ws = (char*)d_ws;
  size_t off = 0;
  float* keys = (float*)(ws + off);   off += (size_t)B * SORT_N * sizeof(float);
  int*   idxs = (int*)(ws + off);     off += (size_t)B * SORT_N * sizeof(int);
  float* boxes = (float*)(ws + off);  off += (size_t)B * NANCH * 4 * sizeof(float);
  unsigned* mask = (unsigned*)(ws + off);
  off += (size_t)B * PRE_NMS * MASK_WORDS * sizeof(unsigned);
  int* keepPos = (int*)(ws + off);    off += (size_t)B * POST_NMS * sizeof(int);
  int* keepCnt = (int*)(ws + off);    off += (size_t)B * sizeof(int);
  (void)ws_size; (void)n_in; (void)in_sizes; (void)out_size;

  // 1) decode + key/index init (incl. pow2 padding)
  {
    int total = B * SORT_N;
    decode_kernel<<<(total + 255) / 256, 256, 0, stream>>>(
        scores, deltas, anchors, im_info, keys, idxs, boxes);
  }

  // 2) descending bitonic sort per 65536-element batch segment
  {
    const int chunks = B * (SORT_N / 1024);
    bitonic_local_sort<<<chunks, 512, 0, stream>>>(keys, idxs);
    for (int k = 2048; k <= SORT_N; k <<= 1) {
      for (int j = k >> 1; j >= 1024; j >>= 1) {
        int total = B * (SORT_N / 2);
        bitonic_global_step<<<(total + 255) / 256, 256, 0, stream>>>(keys, idxs,
                                                                     k, j);
      }
      bitonic_local_merge<<<chunks, 512, 0, stream>>>(keys, idxs, k);
    }
  }

  // 3) suppression bitmask over top-6000 (async-to-LDS column gather)
  {
    dim3 grid((PRE_NMS + 255) / 256, MASK_WORDS, B);
    nms_mask_kernel<<<grid, 256, 0, stream>>>(idxs, boxes, mask);
  }

  // 4) greedy scan (one wave per batch)
  nms_scan_kernel<<<B, 32, 0, stream>>>(keys, mask, keepPos, keepCnt);

  // 5) output
  {
    int total = B * POST_NMS;
    write_out_kernel<<<(total + 127) / 128, 128, 0, stream>>>(idxs, boxes,
                                                              keepPos, keepCnt,
                                                              out);
  }
}